// NUConv2d_85323820302550
// MI455X (gfx1250) — compile-verified
//
#include <hip/hip_runtime.h>

// Problem constants (fixed by the reference harness)
#define BB 4
#define CC 3
#define HH 512
#define WW 512
#define NS 17

// Tiling
#define TILE 32
#define HALO 5                    // |s*mag| < 4.0 strictly, +1 for bilinear corner
#define TDIM (TILE + 2 * HALO)    // 42
#define TSTR (TDIM + 1)           // 43 (odd stride -> fewer LDS bank conflicts)
#define PLANE (TDIM * TSTR)       // 1806 floats per channel plane
#define NTHREADS 256

typedef __attribute__((address_space(1))) int* gas_i32_t;
typedef __attribute__((address_space(3))) int* las_i32_t;

// Async copy of one DWORD from global to LDS (per-lane gather addresses).
// Tracked with ASYNCcnt on gfx1250.
__device__ __forceinline__ void async_copy_b32(const float* g, float* l) {
    float* gnc = const_cast<float*>(g);
    gas_i32_t gp = (gas_i32_t)gnc;
    las_i32_t lp = (las_i32_t)l;
#if __has_builtin(__builtin_amdgcn_global_load_async_to_lds_b32)
    __builtin_amdgcn_global_load_async_to_lds_b32(gp, lp, 0, 0);
#else
    unsigned lds_off = (unsigned)(unsigned long long)lp;
    asm volatile("global_load_async_to_lds_b32 %0, %1, off"
                 :: "v"(lds_off), "v"(gp)
                 : "memory");
#endif
}

__device__ __forceinline__ void wait_async_zero() {
#if __has_builtin(__builtin_amdgcn_s_wait_asynccnt)
    __builtin_amdgcn_s_wait_asynccnt(0);
#else
    asm volatile("s_wait_asynccnt 0x0" ::: "memory");
#endif
}

__global__ __launch_bounds__(NTHREADS)
void nuconv2d_sample_kernel(const float* __restrict__ xin,
                            const float* __restrict__ mag,
                            const float* __restrict__ ori,
                            float* __restrict__ out) {
    __shared__ float tile[CC * PLANE];

    const int bx = blockIdx.x;
    const int by = blockIdx.y;
    const int b  = blockIdx.z;
    const int tid = threadIdx.y * 32 + threadIdx.x;

    const int rowBase = by * TILE - HALO;
    const int colBase = bx * TILE - HALO;

    // ---- Stage the clamped halo tile (3 channels) into LDS via async loads ----
    const int total = CC * TDIM * TDIM; // 5292 elements
    for (int i = tid; i < total; i += NTHREADS) {
        int c   = i / (TDIM * TDIM);
        int rem = i - c * (TDIM * TDIM);
        int r   = rem / TDIM;
        int col = rem - r * TDIM;
        int gr = min(max(rowBase + r,   0), HH - 1);
        int gc = min(max(colBase + col, 0), WW - 1);
        const float* g = xin + (((size_t)b * CC + c) * HH + gr) * WW + gc;
        float* l = &tile[(c * TDIM + r) * TSTR + col];
        async_copy_b32(g, l);
    }
    wait_async_zero();   // this wave's async writes to LDS are done
    __syncthreads();     // all waves' writes visible

    const int x = bx * TILE + threadIdx.x;
    const float fx  = (float)x;
    const float inv = 1.0f / (float)NS;

    const float* t0 = tile;
    const float* t1 = tile + PLANE;
    const float* t2 = tile + 2 * PLANE;

    #pragma unroll
    for (int j = 0; j < 4; ++j) {
        const int y = by * TILE + threadIdx.y + j * 8;
        const float fy = (float)y;

        const size_t midx = ((size_t)b * HH + y) * WW + x;
        const float m = mag[midx];
        const float o = ori[midx];
        float sn, cs;
        __sincosf(o, &sn, &cs);
        const float dx = m * cs;
        const float dy = m * sn;

        float a0 = 0.0f, a1 = 0.0f, a2 = 0.0f;

        for (int k = 0; k < NS; ++k) {
            const float s = -0.5f + (float)k * (1.0f / (float)(NS - 1));
            float sy = fminf(fmaxf(fy + s * dy, 0.0f), (float)(HH - 1));
            float sx = fminf(fmaxf(fx + s * dx, 0.0f), (float)(WW - 1));
            float y0f = floorf(sy);
            float x0f = floorf(sx);
            float wy = sy - y0f;   // == 0.0 exactly when sy is clamped to 511
            float wx = sx - x0f;   // == 0.0 exactly when sx is clamped to 511
            int y0i = (int)y0f;
            int x0i = (int)x0f;

            // Unconditional +1 neighbors: always inside the written 42x42 tile
            // (lx0 <= 40, ly0 <= 40 by |s*mag| < 4); when the reference would
            // clamp x1i/y1i, the corresponding weight is exactly 0.
            const int base = (y0i - rowBase) * TSTR + (x0i - colBase);

            float w00 = (1.0f - wy) * (1.0f - wx);
            float w01 = (1.0f - wy) * wx;
            float w10 = wy * (1.0f - wx);
            float w11 = wy * wx;

            a0 += w00 * t0[base]        + w01 * t0[base + 1]
                + w10 * t0[base + TSTR] + w11 * t0[base + TSTR + 1];
            a1 += w00 * t1[base]        + w01 * t1[base + 1]
                + w10 * t1[base + TSTR] + w11 * t1[base + TSTR + 1];
            a2 += w00 * t2[base]        + w01 * t2[base + 1]
                + w10 * t2[base + TSTR] + w11 * t2[base + TSTR + 1];
        }

        out[(((size_t)b * CC + 0) * HH + y) * WW + x] = a0 * inv;
        out[(((size_t)b * CC + 1) * HH + y) * WW + x] = a1 * inv;
        out[(((size_t)b * CC + 2) * HH + y) * WW + x] = a2 * inv;
    }
}

extern "C" void kernel_launch(void* const* d_in, const int* in_sizes, int n_in,
                              void* d_out, int out_size, void* d_ws, size_t ws_size,
                              hipStream_t stream) {
    (void)in_sizes; (void)n_in; (void)out_size; (void)d_ws; (void)ws_size;
    const float* xin = (const float*)d_in[0];
    const float* mg  = (const float*)d_in[1];
    const float* orn = (const float*)d_in[2];
    float* out = (float*)d_out;

    dim3 grid(WW / TILE, HH / TILE, BB);   // 16 x 16 x 4
    dim3 block(32, 8);                      // 256 threads = 8 wave32 waves
    nuconv2d_sample_kernel<<<grid, block, 0, stream>>>(xin, mg, orn, out);
}